// MultiBoxLoss_9216999817219
// MI455X (gfx1250) — compile-verified
//
#include <hip/hip_runtime.h>
#include <hip/hip_bf16.h>
#include <math.h>

#define BB 16
#define PP 19248
#define CC 81
#define NOBJN 8
#define MDIM 256
#define TILES_PER_B 1203     // PP/16
#define WAVES_K4 8
#define K4_BLK_X 151         // ceil(1203/8)
#define NCH 32
#define CHSZ 602             // ceil(PP/NCH)
#define TILE3 4096

#define LN2F    0.69314718055994530942f
#define LOG2EF  1.44269504088896340736f
#define CLAMP2  (-144.26950408889634f)   // -100 / ln2 (clamp in log2 domain)

typedef float v2f __attribute__((ext_vector_type(2)));
typedef float v8f __attribute__((ext_vector_type(8)));

// ---------------- Kernel 1: matching + smooth-L1 + num_pos -----------------
__global__ __launch_bounds__(512)
void k_match(const float* __restrict__ loc_data, const float* __restrict__ priors,
             const float* __restrict__ gt_boxes, const int* __restrict__ gt_labels,
             float* __restrict__ ws_bto, int* __restrict__ ws_bti,
             int* __restrict__ ws_conf, int* __restrict__ ws_np, float* __restrict__ ws_ll)
{
    __shared__ float s_gt[NOBJN * 4];
    __shared__ int   s_lab[NOBJN];
    __shared__ float s_rf[512];
    __shared__ int   s_ri[512];
    __shared__ int   s_bp[NOBJN];
    const int b   = blockIdx.x;
    const int tid = threadIdx.x;
    if (tid < NOBJN * 4) s_gt[tid]  = gt_boxes[b * NOBJN * 4 + tid];
    if (tid < NOBJN)     s_lab[tid] = gt_labels[b * NOBJN + tid];
    __syncthreads();

    float bestV[NOBJN]; int bestI[NOBJN];
#pragma unroll
    for (int j = 0; j < NOBJN; ++j) { bestV[j] = -1.0f; bestI[j] = PP; }

    const size_t base = (size_t)b * PP;
    for (int p = tid; p < PP; p += 512) {
        const float4 pr = *(const float4*)(priors + (size_t)p * 4);
        const float px1 = pr.x - pr.z * 0.5f, py1 = pr.y - pr.w * 0.5f;
        const float px2 = pr.x + pr.z * 0.5f, py2 = pr.y + pr.w * 0.5f;
        const float areaB = (px2 - px1) * (py2 - py1);
        float bto = -1.0f; int bti = 0;
#pragma unroll
        for (int j = 0; j < NOBJN; ++j) {
            const float gx1 = s_gt[j*4+0], gy1 = s_gt[j*4+1];
            const float gx2 = s_gt[j*4+2], gy2 = s_gt[j*4+3];
            float w = fminf(gx2, px2) - fmaxf(gx1, px1);
            float h = fminf(gy2, py2) - fmaxf(gy1, py1);
            w = fmaxf(w, 0.f); h = fmaxf(h, 0.f);
            const float inter = w * h;
            const float areaA = (gx2 - gx1) * (gy2 - gy1);
            const float iou = inter / (areaA + areaB - inter);
            if (iou > bto)       { bto = iou; bti = j; }           // first-max over objects
            if (iou > bestV[j])  { bestV[j] = iou; bestI[j] = p; } // first-max over priors
        }
        ws_bto[base + p] = bto;
        ws_bti[base + p] = bti;
    }
    __syncthreads();

    // block argmax over P for each object (stable: min index on ties)
    for (int j = 0; j < NOBJN; ++j) {
        s_rf[tid] = bestV[j]; s_ri[tid] = bestI[j];
        __syncthreads();
        for (int s = 256; s > 0; s >>= 1) {
            if (tid < s) {
                const float v2 = s_rf[tid + s]; const int i2 = s_ri[tid + s];
                if (v2 > s_rf[tid] || (v2 == s_rf[tid] && i2 < s_ri[tid])) {
                    s_rf[tid] = v2; s_ri[tid] = i2;
                }
            }
            __syncthreads();
        }
        if (tid == 0) s_bp[j] = s_ri[0];
        __syncthreads();
    }

    // force-match: serial so later objects overwrite (JAX scatter semantics)
    __threadfence_block();
    if (tid == 0) {
        for (int j = 0; j < NOBJN; ++j) {
            const int bp = s_bp[j];
            ws_bto[base + bp] = 2.0f;
            ws_bti[base + bp] = j;
        }
    }
    __threadfence_block();
    __syncthreads();

    // phase 2: conf_t + smooth-L1 over positives
    float llsum = 0.f; int pcount = 0;
    for (int p = tid; p < PP; p += 512) {
        const float bto = ws_bto[base + p];
        const int   j   = ws_bti[base + p];
        int conf = s_lab[j] + 1;
        if (bto < 0.5f) conf = -1;
        if (bto < 0.4f) conf = 0;
        ws_conf[base + p] = conf;
        if (conf > 0) {
            ++pcount;
            const float4 pr = *(const float4*)(priors + (size_t)p * 4);
            const float gx1 = s_gt[j*4+0], gy1 = s_gt[j*4+1];
            const float gx2 = s_gt[j*4+2], gy2 = s_gt[j*4+3];
            const float t0 = (0.5f * (gx1 + gx2) - pr.x) / (0.1f * pr.z);
            const float t1 = (0.5f * (gy1 + gy2) - pr.y) / (0.1f * pr.w);
            const float t2 = logf((gx2 - gx1) / pr.z) / 0.2f;
            const float t3 = logf((gy2 - gy1) / pr.w) / 0.2f;
            const float4 ld = *(const float4*)(loc_data + (base + p) * 4);
            float d;
            d = fabsf(ld.x - t0); llsum += (d < 1.f) ? 0.5f * d * d : d - 0.5f;
            d = fabsf(ld.y - t1); llsum += (d < 1.f) ? 0.5f * d * d : d - 0.5f;
            d = fabsf(ld.z - t2); llsum += (d < 1.f) ? 0.5f * d * d : d - 0.5f;
            d = fabsf(ld.w - t3); llsum += (d < 1.f) ? 0.5f * d * d : d - 0.5f;
        }
    }
    s_rf[tid] = llsum; s_ri[tid] = pcount;
    __syncthreads();
    for (int s = 256; s > 0; s >>= 1) {
        if (tid < s) { s_rf[tid] += s_rf[tid + s]; s_ri[tid] += s_ri[tid + s]; }
        __syncthreads();
    }
    if (tid == 0) { ws_ll[b] = s_rf[0]; ws_np[b] = s_ri[0]; }
}

// ---------------- Kernel 2: per-prior logsumexp -> lc (masked) and ce ------
__global__ __launch_bounds__(256)
void k_conf(const float* __restrict__ conf_data, const int* __restrict__ ws_conf,
            float* __restrict__ ws_lc, float* __restrict__ ws_ce)
{
    const int lane = threadIdx.x & 31;
    const int wid  = blockIdx.x * 8 + (threadIdx.x >> 5);   // prior id, exact (BP % 8 == 0)
    const float* row = conf_data + (size_t)wid * CC;
    const float v0 = row[lane];
    const float v1 = row[lane + 32];
    const bool  h2 = (lane + 64) < CC;
    const float v2 = h2 ? row[lane + 64] : -INFINITY;
    float m = fmaxf(fmaxf(v0, v1), v2);
    for (int o = 16; o >= 1; o >>= 1) m = fmaxf(m, __shfl_xor(m, o, 32));
    // exp(x) = exp2(x*log2e) via bare v_exp_f32; inputs are (v-m) <= 0, finite
    float se = __builtin_amdgcn_exp2f((v0 - m) * LOG2EF)
             + __builtin_amdgcn_exp2f((v1 - m) * LOG2EF)
             + (h2 ? __builtin_amdgcn_exp2f((v2 - m) * LOG2EF) : 0.f);
    for (int o = 16; o >= 1; o >>= 1) se += __shfl_xor(se, o, 32);
    const float lse = m + LN2F * __builtin_amdgcn_logf(se);  // bare v_log_f32
    if (lane == 0) {
        const int ct = ws_conf[wid];
        ws_lc[wid] = (ct != 0) ? 0.f : (lse - row[0]);  // zero for pos and neutral
        const int tgt = (ct > 0) ? ct : 0;
        ws_ce[wid] = lse - row[tgt];                    // -log_softmax[tgt]
    }
}

// ---------------- Kernel 3: exact-rank OHEM + confidence CE sum ------------
__global__ __launch_bounds__(256)
void k_ohem(const float* __restrict__ ws_lc, const float* __restrict__ ws_ce,
            const int* __restrict__ ws_conf, const int* __restrict__ ws_np,
            float* __restrict__ ws_lcp)
{
    __shared__ float s_lc[TILE3];
    __shared__ float s_red[256];
    const int b = blockIdx.y, tid = threadIdx.x;
    const size_t base = (size_t)b * PP;
    int nneg = ws_np[b] * 3; if (nneg > PP - 1) nneg = PP - 1;
    const int c0 = blockIdx.x * CHSZ;
    const int c1 = min(c0 + CHSZ, PP);
    float myv[3]; int myp[3]; int rank[3]; int nl = 0;
    for (int p = c0 + tid; p < c1; p += 256) { myv[nl] = ws_lc[base + p]; myp[nl] = p; rank[nl] = 0; ++nl; }
    for (int t = 0; t < PP; t += TILE3) {
        const int len = min(TILE3, PP - t);
        for (int i = tid; i < len; i += 256) s_lc[i] = ws_lc[base + t + i];
        __syncthreads();
        for (int q = 0; q < len; ++q) {
            const float v = s_lc[q]; const int qq = t + q;
            for (int u = 0; u < nl; ++u)
                rank[u] += (v > myv[u] || (v == myv[u] && qq < myp[u])) ? 1 : 0;
        }
        __syncthreads();
    }
    float csum = 0.f;
    for (int u = 0; u < nl; ++u) {
        const int p  = myp[u];
        const int ct = ws_conf[base + p];
        const bool pos = ct > 0;
        const bool neg = (rank[u] < nneg) && !pos && (ct >= 0);
        if (pos || neg) csum += ws_ce[base + p];
    }
    s_red[tid] = csum; __syncthreads();
    for (int s = 128; s > 0; s >>= 1) { if (tid < s) s_red[tid] += s_red[tid + s]; __syncthreads(); }
    if (tid == 0) ws_lcp[b * NCH + blockIdx.x] = s_red[0];
}

// ---------------- Kernel 4: mask BCE via WMMA f32 16x16x4 (log2 domain) ----
// bce_row(p) = -ln2 * ( [X2 @ gt^T](p, idx_t[p]) + sum_d log2(1-m)(p,d) )
// where X2 = log2(m) - log2(1-m), computed with bare v_log_f32 (no libm fixup)
__global__ __launch_bounds__(256)
void k_mask(const float* __restrict__ mask_data, const float* __restrict__ gt_masks,
            const int* __restrict__ ws_conf, const int* __restrict__ ws_bti,
            float* __restrict__ ws_lmp)
{
    __shared__ float s_gt[16 * MDIM];               // 16 KB: gt_masks[b] padded to 16 rows
    __shared__ float s_G[WAVES_K4][16][16];         // 8 KB: D tiles
    __shared__ float s_ws[WAVES_K4];
    const int b   = blockIdx.y;
    const int tid = threadIdx.x;
    // rows 0..7 = gt_masks[b], rows 8..15 = 0  -> unconditional B fetch, no exec juggling
    for (int i = tid; i < 16 * MDIM; i += 256)
        s_gt[i] = (i < NOBJN * MDIM) ? gt_masks[b * NOBJN * MDIM + i] : 0.f;
    __syncthreads();

    const int wave = __builtin_amdgcn_readfirstlane((int)threadIdx.x) >> 5; // scalar
    const int lane = threadIdx.x & 31;
    const int tile = blockIdx.x * WAVES_K4 + wave;                          // scalar
    float wsum = 0.f;
    if (tile < TILES_PER_B) {   // uniform branch: EXEC stays all-ones for WMMA
        const int half = lane >> 4;
        const int n    = lane & 15;
        const int row  = tile * 16 + n;
        const float* mrow = mask_data + ((size_t)b * PP + row) * MDIM;
        const float* grow = s_gt + n * MDIM;
        v8f c = {0.f, 0.f, 0.f, 0.f, 0.f, 0.f, 0.f, 0.f};
        float s1 = 0.f;
#pragma unroll 4
        for (int k = 0; k < 64; ++k) {
            const int coff = 4 * k + 2 * half;          // A cols / B rows this step
            if ((k & 15) == 0) __builtin_prefetch(mrow + coff + 64, 0, 1);
            const float2 mv = *(const float2*)(mrow + coff);
            // m in (1e-4, 1-1e-4): normals only, clamp never fires; raw v_log_f32 is safe.
            const float lp0 = fmaxf(__builtin_amdgcn_logf(mv.x),        CLAMP2);
            const float lq0 = fmaxf(__builtin_amdgcn_logf(1.0f - mv.x), CLAMP2);
            const float lp1 = fmaxf(__builtin_amdgcn_logf(mv.y),        CLAMP2);
            const float lq1 = fmaxf(__builtin_amdgcn_logf(1.0f - mv.y), CLAMP2);
            s1 += lq0 + lq1;
            v2f a;  a.x = lp0 - lq0;  a.y = lp1 - lq1;  // A[M=n, coff], A[M=n, coff+1]
            v2f bm;                                      // B[K=2*half+{0,1}, N=n]
            bm.x = grow[coff]; bm.y = grow[coff + 1];    // zero rows for n >= 8 (padded)
            c = __builtin_amdgcn_wmma_f32_16x16x4_f32(false, a, false, bm,
                                                      (short)0, c, false, false);
        }
        s1 += __shfl_xor(s1, 16, 32);                   // full 256-col row sum (log2 dom.)
#pragma unroll
        for (int r = 0; r < 8; ++r) s_G[wave][r + 8 * half][n] = c[r];
        float contrib = 0.f;
        if (lane < 16) {
            const int p  = tile * 16 + lane;
            const int ct = ws_conf[(size_t)b * PP + p];
            if (ct > 0) {
                const int j = ws_bti[(size_t)b * PP + p];
                contrib = -LN2F * (s_G[wave][lane][j] + s1);  // back to natural log
            }
        }
        for (int o = 16; o >= 1; o >>= 1) contrib += __shfl_xor(contrib, o, 32);
        wsum = contrib;
    }
    if (lane == 0) s_ws[wave] = wsum;
    __syncthreads();
    if (tid == 0) {
        float s = 0.f;
        for (int w = 0; w < WAVES_K4; ++w) s += s_ws[w];
        ws_lmp[b * K4_BLK_X + blockIdx.x] = s;
    }
}

// ---------------- Kernel 5: deterministic final reduction ------------------
__global__ void k_final(const int* __restrict__ np, const float* __restrict__ ll,
                        const float* __restrict__ lcp, const float* __restrict__ lmp,
                        float* __restrict__ out)
{
    if (threadIdx.x == 0 && blockIdx.x == 0) {
        float N = 0.f, a = 0.f, c = 0.f, m = 0.f;
        for (int b = 0; b < BB; ++b) { N += (float)np[b]; a += ll[b]; }
        for (int i = 0; i < BB * NCH; ++i)      c += lcp[i];
        for (int i = 0; i < BB * K4_BLK_X; ++i) m += lmp[i];
        out[0] = a / N;                          // BBOX_ALPHA = 1.0
        out[1] = c / N;
        out[2] = m * (0.2f / 256.0f) / N;        // MASK_ALPHA = 0.2/MD
    }
}

extern "C" void kernel_launch(void* const* d_in, const int* in_sizes, int n_in,
                              void* d_out, int out_size, void* d_ws, size_t ws_size,
                              hipStream_t stream) {
    const float* loc    = (const float*)d_in[0];
    const float* conf   = (const float*)d_in[1];
    const float* mask   = (const float*)d_in[2];
    const float* priors = (const float*)d_in[3];
    const float* gtb    = (const float*)d_in[4];
    const int*   gtl    = (const int*)  d_in[5];
    const float* gtm    = (const float*)d_in[6];
    float* out = (float*)d_out;

    const size_t BP = (size_t)BB * PP;
    float* ws_bto  = (float*)d_ws;                // [B*P]
    int*   ws_bti  = (int*)(ws_bto + BP);         // [B*P]
    int*   ws_conf = ws_bti + BP;                 // [B*P]
    float* ws_lc   = (float*)(ws_conf + BP);      // [B*P]
    float* ws_ce   = ws_lc + BP;                  // [B*P]
    int*   ws_np   = (int*)(ws_ce + BP);          // [B]
    float* ws_ll   = (float*)(ws_np + BB);        // [B]
    float* ws_lcp  = ws_ll + BB;                  // [B*NCH]
    float* ws_lmp  = ws_lcp + BB * NCH;           // [B*K4_BLK_X]

    hipLaunchKernelGGL(k_match, dim3(BB), dim3(512), 0, stream,
                       loc, priors, gtb, gtl, ws_bto, ws_bti, ws_conf, ws_np, ws_ll);
    hipLaunchKernelGGL(k_conf, dim3((unsigned)(BP / 8)), dim3(256), 0, stream,
                       conf, ws_conf, ws_lc, ws_ce);
    hipLaunchKernelGGL(k_ohem, dim3(NCH, BB), dim3(256), 0, stream,
                       ws_lc, ws_ce, ws_conf, ws_np, ws_lcp);
    hipLaunchKernelGGL(k_mask, dim3(K4_BLK_X, BB), dim3(256), 0, stream,
                       mask, gtm, ws_conf, ws_bti, ws_lmp);
    hipLaunchKernelGGL(k_final, dim3(1), dim3(32), 0, stream,
                       ws_np, ws_ll, ws_lcp, ws_lmp, out);
}